// DHGNNRawConv_60335700574234
// MI455X (gfx1250) — compile-verified
//
#include <hip/hip_runtime.h>
#include <hip/hip_bf16.h>

#define D_IN   128
#define D_OUT  256
#define K_S    16
#define K_K    8

typedef __attribute__((ext_vector_type(2))) float v2f;
typedef __attribute__((ext_vector_type(8))) float v8f;

// ---------------------------------------------------------------------------
// Kernel A: h = x @ W   (M=50000, K=128, N=256), fp32 WMMA 16x16x4.
// One wave computes a 16(M) x 64(N) strip: 4 column tiles, K-loop of 32 steps.
// Block = 256 threads = 8 waves -> 32 rows x 256 cols per block.
//
// VGPR layouts per CDNA5 ISA 7.12.2:
//   A (16x4 f32): lanes 0-15 hold M=lane, K={0,1} in v0,v1; lanes 16-31 K={2,3}.
//   B (4x16 f32): v0 = rows K=0 (lanes 0-15) / K=2 (lanes 16-31); v1 = K=1/K=3.
//   C/D (16x16 f32): VGPR r = row r (lanes 0-15) / row r+8 (lanes 16-31).
// ---------------------------------------------------------------------------
__global__ __launch_bounds__(256) void dhgnn_gemm_h(const float* __restrict__ x,
                                                    const float* __restrict__ W,
                                                    float* __restrict__ h,
                                                    int N) {
    const int lane = threadIdx.x & 31;
    const int wave = threadIdx.x >> 5;
    const int rowTile = blockIdx.x * 2 + (wave >> 2);
    const int row0 = rowTile * 16;
    if (row0 >= N) return;                       // wave-uniform exit
    const int col0 = (wave & 3) * 64;

    const int m    = lane & 15;                  // row/col within tile
    const int koff = (lane >> 4) << 1;           // 0 for lanes 0-15, 2 for 16-31

    int rowA = row0 + m;
    if (rowA >= N) rowA = N - 1;                 // clamp (stores are predicated)
    const float* __restrict__ xrow = x + (size_t)rowA * D_IN;

    v8f c0 = {}, c1 = {}, c2 = {}, c3 = {};

    for (int k = 0; k < D_IN; k += 4) {
        v2f a;
        a.x = xrow[k + koff + 0];
        a.y = xrow[k + koff + 1];

        const float* __restrict__ wk = W + (size_t)(k + koff) * D_OUT + m;
        v2f b0, b1, b2, b3;
        b0.x = wk[col0 +  0]; b0.y = wk[D_OUT + col0 +  0];
        b1.x = wk[col0 + 16]; b1.y = wk[D_OUT + col0 + 16];
        b2.x = wk[col0 + 32]; b2.y = wk[D_OUT + col0 + 32];
        b3.x = wk[col0 + 48]; b3.y = wk[D_OUT + col0 + 48];

        c0 = __builtin_amdgcn_wmma_f32_16x16x4_f32(false, a, false, b0, (short)0, c0, false, false);
        c1 = __builtin_amdgcn_wmma_f32_16x16x4_f32(false, a, false, b1, (short)0, c1, false, false);
        c2 = __builtin_amdgcn_wmma_f32_16x16x4_f32(false, a, false, b2, (short)0, c2, false, false);
        c3 = __builtin_amdgcn_wmma_f32_16x16x4_f32(false, a, false, b3, (short)0, c3, false, false);
    }

    const int mrow = row0 + ((lane >> 4) << 3);  // +8 row offset for upper lanes
    const int ccol = col0 + m;
#pragma unroll
    for (int r = 0; r < 8; ++r) {
        const int rr = mrow + r;
        if (rr < N) {
            float* __restrict__ hr = h + (size_t)rr * D_OUT + ccol;
            hr[ 0] = c0[r];
            hr[16] = c1[r];
            hr[32] = c2[r];
            hr[48] = c3[r];
        }
    }
}

// ---------------------------------------------------------------------------
// Kernel B: fused per-node branch. One block (256 thr = 8 waves) per node.
//  - gather 16x256 (edge) and 8x256 (knn) rows of h into LDS (L2-resident)
//  - depthwise conv -> per-row softmax -> coef[j] = sum_i wK1[i]*mult[i][j]
//  - pooled[d] = bK1 + sum_j coef[j] * region[j][d]
//  - out = pooled_s + pooled_k + bias   (attention softmax over singleton == 1)
// ---------------------------------------------------------------------------
__global__ __launch_bounds__(256) void dhgnn_fused(const float* __restrict__ h,
                                                   const int*   __restrict__ eidx,
                                                   const int*   __restrict__ kidx,
                                                   const float* __restrict__ wKKs,  // (256,1,16)
                                                   const float* __restrict__ bKKs,  // (256)
                                                   const float* __restrict__ wK1s,  // (1,16,1)
                                                   const float* __restrict__ bK1s,  // (1)
                                                   const float* __restrict__ wKKk,  // (64,4,8)
                                                   const float* __restrict__ bKKk,  // (64)
                                                   const float* __restrict__ wK1k,  // (1,8,1)
                                                   const float* __restrict__ bK1k,  // (1)
                                                   const float* __restrict__ bias,  // (256)
                                                   float* __restrict__ out,
                                                   int N) {
    __shared__ float Rs[K_S * D_OUT];   // 16 KB gathered edge region
    __shared__ float Rk[K_K * D_OUT];   //  8 KB gathered knn region
    __shared__ float sm[D_OUT];         // conv / softmax scratch
    __shared__ float coef[K_S];         // collapsed pooling coefficients
    __shared__ int   nbrS[K_S];
    __shared__ int   nbrK[K_K];

    const int n   = blockIdx.x;
    const int tid = threadIdx.x;
    if (n >= N) return;

    if (tid < K_S) nbrS[tid] = eidx[(size_t)n * K_S + tid];
    if (tid < K_K) nbrK[tid] = kidx[(size_t)n * K_K + tid];
    __syncthreads();

    // ---- gather regions (float4, coalesced; rows of h are L2-resident) ----
    for (int idx = tid; idx < K_S * (D_OUT / 4); idx += 256) {
        const int r = idx >> 6, c = (idx & 63) << 2;
        *reinterpret_cast<float4*>(&Rs[r * D_OUT + c]) =
            *reinterpret_cast<const float4*>(h + (size_t)nbrS[r] * D_OUT + c);
    }
    for (int idx = tid; idx < K_K * (D_OUT / 4); idx += 256) {
        const int r = idx >> 6, c = (idx & 63) << 2;
        *reinterpret_cast<float4*>(&Rk[r * D_OUT + c]) =
            *reinterpret_cast<const float4*>(h + (size_t)nbrK[r] * D_OUT + c);
    }
    __syncthreads();

    // ================= edge (s) branch, 16x16 =================
    {   // depthwise conv: conved[o] = bKKs[o] + sum_k Rs[k][o]*wKKs[o][k]
        float acc = bKKs[tid];
#pragma unroll
        for (int k = 0; k < K_S; ++k)
            acc += Rs[k * D_OUT + tid] * wKKs[tid * K_S + k];
        sm[tid] = acc;
    }
    __syncthreads();
    {   // row softmax over j (o = i*16+j), redundantly per thread of row
        const int i = tid >> 4;
        float v[K_S];
#pragma unroll
        for (int j = 0; j < K_S; ++j) v[j] = sm[i * K_S + j];
        float mx = v[0];
#pragma unroll
        for (int j = 1; j < K_S; ++j) mx = fmaxf(mx, v[j]);
        float s = 0.f;
#pragma unroll
        for (int j = 0; j < K_S; ++j) s += __expf(v[j] - mx);
        const float mine = __expf(v[tid & 15] - mx) / s;
        __syncthreads();
        sm[tid] = mine;                        // mult[i][j]
    }
    __syncthreads();
    if (tid < K_S) {                           // coef[j] = sum_i wK1s[i]*mult[i][j]
        float c = 0.f;
#pragma unroll
        for (int i = 0; i < K_S; ++i) c += wK1s[i] * sm[i * K_S + tid];
        coef[tid] = c;
    }
    __syncthreads();
    float pooled_s = bK1s[0];
#pragma unroll
    for (int j = 0; j < K_S; ++j) pooled_s += coef[j] * Rs[j * D_OUT + tid];
    __syncthreads();                            // done reading sm/coef

    // ================= knn (k) branch, 8x8, groups=64 =================
    if (tid < K_K * K_K) {                      // conved[o] = b + sum_i sum_kk Rk[kk][4o+i]*w[o][i][kk]
        float acc = bKKk[tid];
#pragma unroll
        for (int i = 0; i < 4; ++i)
#pragma unroll
            for (int kk = 0; kk < K_K; ++kk)
                acc += Rk[kk * D_OUT + tid * 4 + i] * wKKk[tid * 32 + i * 8 + kk];
        sm[tid] = acc;
    }
    __syncthreads();
    if (tid < K_K * K_K) {
        const int i = tid >> 3;
        float v[K_K];
#pragma unroll
        for (int j = 0; j < K_K; ++j) v[j] = sm[i * K_K + j];
        float mx = v[0];
#pragma unroll
        for (int j = 1; j < K_K; ++j) mx = fmaxf(mx, v[j]);
        float s = 0.f;
#pragma unroll
        for (int j = 0; j < K_K; ++j) s += __expf(v[j] - mx);
        const float mine = __expf(v[tid & 7] - mx) / s;
        __syncthreads();
        sm[tid] = mine;
    } else {
        __syncthreads();
    }
    __syncthreads();
    if (tid < K_K) {
        float c = 0.f;
#pragma unroll
        for (int i = 0; i < K_K; ++i) c += wK1k[i] * sm[i * K_K + tid];
        coef[tid] = c;
    }
    __syncthreads();
    float pooled_k = bK1k[0];
#pragma unroll
    for (int j = 0; j < K_K; ++j) pooled_k += coef[j] * Rk[j * D_OUT + tid];

    // attention softmax over singleton dim == 1 -> out = x_s + x_k + bias
    out[(size_t)n * D_OUT + tid] = pooled_s + pooled_k + bias[tid];
}

extern "C" void kernel_launch(void* const* d_in, const int* in_sizes, int n_in,
                              void* d_out, int out_size, void* d_ws, size_t ws_size,
                              hipStream_t stream) {
    const float* x    = (const float*)d_in[0];
    const int*   eidx = (const int*)  d_in[1];
    const int*   kidx = (const int*)  d_in[2];
    const float* W    = (const float*)d_in[3];
    const float* bias = (const float*)d_in[4];
    const float* wKKs = (const float*)d_in[5];
    const float* bKKs = (const float*)d_in[6];
    const float* wK1s = (const float*)d_in[7];
    const float* bK1s = (const float*)d_in[8];
    const float* wKKk = (const float*)d_in[9];
    const float* bKKk = (const float*)d_in[10];
    const float* wK1k = (const float*)d_in[11];
    const float* bK1k = (const float*)d_in[12];
    // d_in[13..16]: attention MLP — mathematically dead (softmax over dim of size 1)

    float* out = (float*)d_out;
    float* h   = (float*)d_ws;                 // 50000*256*4 = 51.2 MB scratch

    const int N = in_sizes[0] / D_IN;

    const int rowTiles = (N + 15) / 16;
    const int blocksA  = (rowTiles + 1) / 2;   // 2 row-tiles per block
    dhgnn_gemm_h<<<blocksA, 256, 0, stream>>>(x, W, h, N);

    dhgnn_fused<<<N, 256, 0, stream>>>(h, eidx, kidx,
                                       wKKs, bKKs, wK1s, bK1s,
                                       wKKk, bKKk, wK1k, bK1k,
                                       bias, out, N);
}